// Encoder_72971494359297
// MI455X (gfx1250) — compile-verified
//
#include <hip/hip_runtime.h>
#include <hip/hip_bf16.h>

typedef __attribute__((ext_vector_type(2))) float v2f;
typedef __attribute__((ext_vector_type(8))) float v8f;

// ---------------------------------------------------------------------------
// CSR construction: histogram -> single-block scan (+dinv) -> cursor fill
// ---------------------------------------------------------------------------
__global__ void zero_cnt_k(int* __restrict__ cnt, int n) {
    int i = blockIdx.x * blockDim.x + threadIdx.x;
    if (i < n) cnt[i] = 0;
}

__global__ void count_k(const int* __restrict__ ei, int* __restrict__ cnt, int E) {
    int e = blockIdx.x * blockDim.x + threadIdx.x;
    if (e < E) atomicAdd(&cnt[ei[E + e]], 1);            // in-degree over dst
}

// Exclusive scan of cnt[0..n) into row_ptr[0..n], init cursor, and
// dinv[i] = rsqrt(deg) with deg = cnt+1 (self-loop). Single 1024-thread block.
__global__ void scan_dinv_k(const int* __restrict__ cnt, int* __restrict__ row_ptr,
                            int* __restrict__ cursor, float* __restrict__ dinv, int n) {
    __shared__ int sdata[1024];
    __shared__ int running;
    if (threadIdx.x == 0) running = 0;
    __syncthreads();
    for (int base = 0; base < n; base += 1024) {
        int i = base + (int)threadIdx.x;
        int v = (i < n) ? cnt[i] : 0;
        sdata[threadIdx.x] = v;
        __syncthreads();
        for (int off = 1; off < 1024; off <<= 1) {
            int t = (threadIdx.x >= (unsigned)off) ? sdata[threadIdx.x - off] : 0;
            __syncthreads();
            sdata[threadIdx.x] += t;
            __syncthreads();
        }
        int incl = sdata[threadIdx.x];
        int base_off = running;
        if (i < n) {
            int excl = base_off + incl - v;
            row_ptr[i] = excl;
            cursor[i]  = excl;
            dinv[i]    = rsqrtf((float)(v + 1));
        }
        __syncthreads();
        if (threadIdx.x == 1023) running = base_off + incl;
        __syncthreads();
    }
    if (threadIdx.x == 0) row_ptr[n] = running;
}

__global__ void fill_k(const int* __restrict__ ei, int* __restrict__ cursor,
                       int* __restrict__ csr_src, int E) {
    int e = blockIdx.x * blockDim.x + threadIdx.x;
    if (e >= E) return;
    int d   = ei[E + e];
    int pos = atomicAdd(&cursor[d], 1);
    csr_src[pos] = ei[e];
}

// ---------------------------------------------------------------------------
// FP32 WMMA GEMM: C[M,Nc] = A[M,K] @ W[K,Nc]
// Each wave computes an 80x16 output panel: 5 m-subtiles of 16x16 sharing one
// B operand per k-step (M/16 = 3125 = 5*625, exact -> EXEC stays all-1s).
// Explicit 2-stage software pipeline; __launch_bounds__(256,1) lifts the
// occupancy-driven VGPR cap.
// A operand (16x4):  lane&15 = M row, lane>>4 selects K pair {0,1}/{2,3}
// B operand (4x16):  lane&15 = N col, lane>>4 selects K pair; v0=K even, v1=K odd
// C/D (16x16):       VGPR v -> M = v + 8*(lane>>4), N = lane&15
// ---------------------------------------------------------------------------
__global__ void __launch_bounds__(256, 1)
gemm_wmma_f32(const float* __restrict__ A, const float* __restrict__ W,
              float* __restrict__ C, int M, int K, int Nc) {
    const int wave   = (blockIdx.x * blockDim.x + threadIdx.x) >> 5;
    const int lane   = threadIdx.x & 31;
    const int tilesN = Nc >> 4;
    const int group  = wave / tilesN;
    const int tile_n = wave % tilesN;
    const int m0     = group * 80;
    if (m0 >= M) return;                       // wave-uniform guard

    const int row  = lane & 15;
    const int half = lane >> 4;

    const float* a_base = A + (size_t)(m0 + row) * K + half * 2;
    const float* b_ptr  = W + (size_t)(half * 2) * Nc + tile_n * 16 + row;
    const size_t aSub   = (size_t)16 * K;
    const size_t bStep  = (size_t)4 * Nc;

    v8f acc0 = {}, acc1 = {}, acc2 = {}, acc3 = {}, acc4 = {};

    v2f bc, a0c, a1c, a2c, a3c, a4c;
    bc.x = b_ptr[0];
    bc.y = b_ptr[Nc];
    a0c = *(const v2f*)(a_base + 0 * aSub);
    a1c = *(const v2f*)(a_base + 1 * aSub);
    a2c = *(const v2f*)(a_base + 2 * aSub);
    a3c = *(const v2f*)(a_base + 3 * aSub);
    a4c = *(const v2f*)(a_base + 4 * aSub);

    for (int k0 = 0; k0 < K - 4; k0 += 4) {
        const float* a_nxt = a_base + 4;
        const float* b_nxt = b_ptr + bStep;
        v2f bn, a0n, a1n, a2n, a3n, a4n;
        bn.x = b_nxt[0];
        bn.y = b_nxt[Nc];
        a0n = *(const v2f*)(a_nxt + 0 * aSub);
        a1n = *(const v2f*)(a_nxt + 1 * aSub);
        a2n = *(const v2f*)(a_nxt + 2 * aSub);
        a3n = *(const v2f*)(a_nxt + 3 * aSub);
        a4n = *(const v2f*)(a_nxt + 4 * aSub);

        acc0 = __builtin_amdgcn_wmma_f32_16x16x4_f32(false, a0c, false, bc, (short)0, acc0, false, false);
        acc1 = __builtin_amdgcn_wmma_f32_16x16x4_f32(false, a1c, false, bc, (short)0, acc1, false, false);
        acc2 = __builtin_amdgcn_wmma_f32_16x16x4_f32(false, a2c, false, bc, (short)0, acc2, false, false);
        acc3 = __builtin_amdgcn_wmma_f32_16x16x4_f32(false, a3c, false, bc, (short)0, acc3, false, false);
        acc4 = __builtin_amdgcn_wmma_f32_16x16x4_f32(false, a4c, false, bc, (short)0, acc4, false, false);

        bc = bn; a0c = a0n; a1c = a1n; a2c = a2n; a3c = a3n; a4c = a4n;
        a_base = a_nxt;
        b_ptr  = b_nxt;
    }

    acc0 = __builtin_amdgcn_wmma_f32_16x16x4_f32(false, a0c, false, bc, (short)0, acc0, false, false);
    acc1 = __builtin_amdgcn_wmma_f32_16x16x4_f32(false, a1c, false, bc, (short)0, acc1, false, false);
    acc2 = __builtin_amdgcn_wmma_f32_16x16x4_f32(false, a2c, false, bc, (short)0, acc2, false, false);
    acc3 = __builtin_amdgcn_wmma_f32_16x16x4_f32(false, a3c, false, bc, (short)0, acc3, false, false);
    acc4 = __builtin_amdgcn_wmma_f32_16x16x4_f32(false, a4c, false, bc, (short)0, acc4, false, false);

    float* c_base = C + (size_t)(m0 + half * 8) * Nc + tile_n * 16 + row;
    const size_t cSub = (size_t)16 * Nc;
#pragma unroll
    for (int v = 0; v < 8; ++v) {
        c_base[(size_t)v * Nc + 0 * cSub] = acc0[v];
        c_base[(size_t)v * Nc + 1 * cSub] = acc1[v];
        c_base[(size_t)v * Nc + 2 * cSub] = acc2[v];
        c_base[(size_t)v * Nc + 3 * cSub] = acc3[v];
        c_base[(size_t)v * Nc + 4 * cSub] = acc4[v];
    }
}

// ---------------------------------------------------------------------------
// Fused CSR gather-aggregate + self-loop + bias (+ ReLU):
//   out[i,c] = act( dinv[i] * sum_j dinv[src_j] * h[src_j,c]
//                 + dinv[i]^2 * h[i,c] + bias[c] )
// One wave owns one node's channel slice: edge loop is wave-uniform (no
// divergence), csr_src[j] is a same-address broadcast load, h gathers are
// coalesced float4 and L2-resident. Zero floating-point atomics.
// ---------------------------------------------------------------------------
template <int C, bool RELU>
__global__ void aggregate_k(const float* __restrict__ h, const int* __restrict__ row_ptr,
                            const int* __restrict__ csr_src, const float* __restrict__ dinv,
                            const float* __restrict__ bias, float* __restrict__ out, int n) {
    constexpr int lanesPerNode = C / 4;        // 64 (C=256) or 32 (C=128)
    int t = blockIdx.x * blockDim.x + threadIdx.x;
    int i = t / lanesPerNode;
    if (i >= n) return;
    int c = (t % lanesPerNode) * 4;

    int beg = row_ptr[i];
    int end = row_ptr[i + 1];
    float di = dinv[i];

    float ax = 0.f, ay = 0.f, az = 0.f, aw = 0.f;
    for (int j = beg; j < end; ++j) {
        int s   = csr_src[j];                  // wave-uniform broadcast
        float w = dinv[s];
        float4 hv = *(const float4*)(h + (size_t)s * C + c);
        ax = fmaf(w, hv.x, ax);
        ay = fmaf(w, hv.y, ay);
        az = fmaf(w, hv.z, az);
        aw = fmaf(w, hv.w, aw);
    }

    float4 hs = *(const float4*)(h + (size_t)i * C + c);
    float  ws = di * di;                       // self-loop weight
    float4 bv = *(const float4*)(bias + c);
    float4 r;
    r.x = fmaf(di, ax, fmaf(ws, hs.x, bv.x));
    r.y = fmaf(di, ay, fmaf(ws, hs.y, bv.y));
    r.z = fmaf(di, az, fmaf(ws, hs.z, bv.z));
    r.w = fmaf(di, aw, fmaf(ws, hs.w, bv.w));
    if (RELU) {
        r.x = fmaxf(r.x, 0.f);
        r.y = fmaxf(r.y, 0.f);
        r.z = fmaxf(r.z, 0.f);
        r.w = fmaxf(r.w, 0.f);
    }
    *(float4*)(out + (size_t)i * C + c) = r;
}

// ---------------------------------------------------------------------------
// Orchestration
// ---------------------------------------------------------------------------
extern "C" void kernel_launch(void* const* d_in, const int* in_sizes, int n_in,
                              void* d_out, int out_size, void* d_ws, size_t ws_size,
                              hipStream_t stream) {
    const float* x  = (const float*)d_in[0];
    const int*   ei = (const int*)d_in[1];
    const float* W1 = (const float*)d_in[2];
    const float* b1 = (const float*)d_in[3];
    const float* W2 = (const float*)d_in[4];
    const float* b2 = (const float*)d_in[5];
    float* out = (float*)d_out;

    const int N  = 50000;
    const int E  = in_sizes[1] / 2;    // edge_index is [2, E]
    const int K1 = 256, C1 = 256, C2 = 128;

    // Workspace layout: h1 | h1a | h2 | dinv | cnt | row_ptr | cursor | csr_src
    float* h1   = (float*)d_ws;
    float* h1a  = h1  + (size_t)N * C1;
    float* h2   = h1a + (size_t)N * C1;
    float* dinv = h2  + (size_t)N * C2;
    int*   cnt     = (int*)(dinv + N);
    int*   row_ptr = cnt + N;          // N+1 entries
    int*   cursor  = row_ptr + (N + 1);
    int*   csr_src = cursor + N;       // E entries

    const int TB = 256;
    const int groupsM = (N + 79) / 80; // 80-row panels per GEMM wave (625 exact)

    // 1) CSR build + normalization (reused by both layers)
    zero_cnt_k <<<(N + TB - 1) / TB, TB, 0, stream>>>(cnt, N);
    count_k    <<<(E + TB - 1) / TB, TB, 0, stream>>>(ei, cnt, E);
    scan_dinv_k<<<1, 1024, 0, stream>>>(cnt, row_ptr, cursor, dinv, N);
    fill_k     <<<(E + TB - 1) / TB, TB, 0, stream>>>(ei, cursor, csr_src, E);

    // 2) layer 1: h1 = x @ W1 ; h1a = relu(gather-aggregate + self + b1)
    {
        int waves  = groupsM * (C1 / 16);
        int blocks = (waves * 32 + TB - 1) / TB;
        gemm_wmma_f32<<<blocks, TB, 0, stream>>>(x, W1, h1, N, K1, C1);
    }
    {
        long long work = (long long)N * (C1 / 4);
        aggregate_k<C1, true><<<(unsigned)((work + TB - 1) / TB), TB, 0, stream>>>(
            h1, row_ptr, csr_src, dinv, b1, h1a, N);
    }

    // 3) layer 2: h2 = h1a @ W2 ; out = gather-aggregate + self + b2
    {
        int waves  = groupsM * (C2 / 16);
        int blocks = (waves * 32 + TB - 1) / TB;
        gemm_wmma_f32<<<blocks, TB, 0, stream>>>(h1a, W2, h2, N, C1, C2);
    }
    {
        long long work = (long long)N * (C2 / 4);
        aggregate_k<C2, false><<<(unsigned)((work + TB - 1) / TB), TB, 0, stream>>>(
            h2, row_ptr, csr_src, dinv, b2, out, N);
    }
}